// BesselFilter_3599182594850
// MI455X (gfx1250) — compile-verified
//
#include <hip/hip_runtime.h>
#include <hip/hip_bf16.h>

// ---------------- problem constants ----------------
#define T_LEN   (1u << 20)          // 1048576 samples per row
#define BATCH   16
#define CHUNK_L 2048                // chunk length (2^11)
#define LOG2_L  11
#define CPR     512                 // chunks per row = T_LEN / CHUNK_L
#define NCHUNK  (BATCH * CPR)       // 8192 total chunks
#define NSTATE  5                   // filter order

// ---------------- workspace layout (float offsets) ----------------
#define OFF_MAXBITS 0               // 1 uint  : max |x| as float bits
#define OFF_P       64              // 25 f32  : P = M^CHUNK_L (row major, z' = P z)
#define OFF_H       1024            // CHUNK_L*8 f32 : h(t) rows padded to 8 (cols 5..7 = 0)
#define OFF_F       20480           // NCHUNK*8 f32  : zero-state final state per chunk (padded)
#define OFF_Z       90112           // BATCH*8*CPR f32: z_start, component-major per row (padded)
// total ~155648 floats ~ 608 KB of d_ws

typedef __attribute__((ext_vector_type(2))) float v2f;
typedef __attribute__((ext_vector_type(8))) float v8f;

// ---------------- kernel 0: init max slot ----------------
__global__ void k_init(unsigned* wsu) {
    if (threadIdx.x == 0 && blockIdx.x == 0) wsu[OFF_MAXBITS] = 0u;
}

// ---------------- kernel 1: abs-max reduction ----------------
__global__ void __launch_bounds__(256) k_absmax(const float* __restrict__ x,
                                                unsigned* __restrict__ wsu, int n) {
    __shared__ unsigned sm[256];
    unsigned m = 0u;
    for (int i = blockIdx.x * blockDim.x + threadIdx.x; i < n;
         i += gridDim.x * blockDim.x) {
        unsigned b = __float_as_uint(fabsf(x[i]));   // non-negative: bit order == float order
        m = (b > m) ? b : m;
    }
    sm[threadIdx.x] = m;
    __syncthreads();
    for (int s = 128; s > 0; s >>= 1) {
        if ((int)threadIdx.x < s) {
            unsigned o = sm[threadIdx.x + s];
            if (o > sm[threadIdx.x]) sm[threadIdx.x] = o;
        }
        __syncthreads();
    }
    if (threadIdx.x == 0) atomicMax(&wsu[OFF_MAXBITS], sm[0]);
}

// ---------------- kernel 2: prep P = M^L and H (homogeneous responses) ----------------
__global__ void k_prep(const float* __restrict__ bc, const float* __restrict__ ac,
                       float* __restrict__ ws) {
    if (threadIdx.x != 0 || blockIdx.x != 0) return;
    float ia0 = 1.0f / ac[0];
    float a[6];
    #pragma unroll
    for (int i = 0; i < 6; ++i) a[i] = ac[i] * ia0;

    // DF2T companion matrix: z'(i) = -a[i+1]*z0 + z(i+1)   (z5 == 0)
    float M[25];
    #pragma unroll
    for (int i = 0; i < 25; ++i) M[i] = 0.0f;
    #pragma unroll
    for (int i = 0; i < NSTATE; ++i) {
        M[i * NSTATE + 0] = -a[i + 1];
        if (i < NSTATE - 1) M[i * NSTATE + (i + 1)] += 1.0f;
    }

    // P = M^(2^LOG2_L) by repeated squaring
    float P[25], Q[25];
    #pragma unroll
    for (int i = 0; i < 25; ++i) P[i] = M[i];
    for (int s = 0; s < LOG2_L; ++s) {
        for (int i = 0; i < NSTATE; ++i)
            for (int j = 0; j < NSTATE; ++j) {
                float acc = 0.0f;
                for (int k = 0; k < NSTATE; ++k)
                    acc = fmaf(P[i * NSTATE + k], P[k * NSTATE + j], acc);
                Q[i * NSTATE + j] = acc;
            }
        for (int i = 0; i < 25; ++i) P[i] = Q[i];
    }
    for (int i = 0; i < 25; ++i) ws[OFF_P + i] = P[i];

    // H rows: h(0) = e0 ; h(t+1) = [ -sum a[k+1] h[k], h0, h1, h2, h3 ]
    float h0 = 1.0f, h1 = 0.0f, h2 = 0.0f, h3 = 0.0f, h4 = 0.0f;
    float* H = ws + OFF_H;
    for (int t = 0; t < CHUNK_L; ++t) {
        float* r = H + (size_t)t * 8;
        r[0] = h0; r[1] = h1; r[2] = h2; r[3] = h3; r[4] = h4;
        r[5] = 0.0f; r[6] = 0.0f; r[7] = 0.0f;
        float nh0 = -(fmaf(a[1], h0, fmaf(a[2], h1, fmaf(a[3], h2, fmaf(a[4], h3, a[5] * h4)))));
        h4 = h3; h3 = h2; h2 = h1; h1 = h0; h0 = nh0;
    }
}

// ---------------- kernel 3: per-chunk zero-state filtering ----------------
__global__ void __launch_bounds__(256) k_phase1(const float* __restrict__ x,
                                                const float* __restrict__ bc,
                                                const float* __restrict__ ac,
                                                float* __restrict__ out,
                                                float* __restrict__ ws) {
    int tid = blockIdx.x * blockDim.x + threadIdx.x;
    if (tid >= NCHUNK) return;
    int row = tid / CPR;
    int c   = tid % CPR;

    float xmax = __uint_as_float(((const unsigned*)ws)[OFF_MAXBITS]);
    float inv  = 1.0f / xmax;

    float ia0 = 1.0f / ac[0];
    float b0 = bc[0] * ia0, b1 = bc[1] * ia0, b2 = bc[2] * ia0,
          b3 = bc[3] * ia0, b4 = bc[4] * ia0, b5 = bc[5] * ia0;
    float a1 = ac[1] * ia0, a2 = ac[2] * ia0, a3 = ac[3] * ia0,
          a4 = ac[4] * ia0, a5 = ac[5] * ia0;

    size_t base = (size_t)row * T_LEN + (size_t)c * CHUNK_L;
    const float* xp = x + base;
    float*       op = out + base;

    float z0 = 0.f, z1 = 0.f, z2 = 0.f, z3 = 0.f, z4 = 0.f;
    for (int t = 0; t < CHUNK_L; ++t) {
        if ((t & 31) == 0) __builtin_prefetch(xp + t + 128, 0, 0);  // global_prefetch_b8
        float xt = xp[t] * inv;
        float y  = fmaf(b0, xt, z0);
        z0 = fmaf(-a1, y, fmaf(b1, xt, z1));
        z1 = fmaf(-a2, y, fmaf(b2, xt, z2));
        z2 = fmaf(-a3, y, fmaf(b3, xt, z3));
        z3 = fmaf(-a4, y, fmaf(b4, xt, z4));
        z4 = fmaf(-a5, y, b5 * xt);
        op[t] = y;                         // normalized zero-state output
    }
    float* f = ws + OFF_F + (size_t)tid * 8;
    f[0] = z0; f[1] = z1; f[2] = z2; f[3] = z3; f[4] = z4;
    f[5] = 0.f; f[6] = 0.f; f[7] = 0.f;
}

// ---------------- kernel 4: sequential chunk-state scan (per row) ----------------
__global__ void k_phase2(float* __restrict__ ws) {
    int r = threadIdx.x;
    if (r >= BATCH || blockIdx.x != 0) return;
    float P[25];
    #pragma unroll
    for (int i = 0; i < 25; ++i) P[i] = ws[OFF_P + i];

    float z[NSTATE] = {0.f, 0.f, 0.f, 0.f, 0.f};
    float*       Zr = ws + OFF_Z + (size_t)r * 8 * CPR;   // Zr[k*CPR + c]
    const float* fr = ws + OFF_F + (size_t)r * CPR * 8;   // fr[c*8 + k]

    for (int c = 0; c < CPR; ++c) {
        #pragma unroll
        for (int k = 0; k < NSTATE; ++k) Zr[k * CPR + c] = z[k];  // z_start for chunk c
        #pragma unroll
        for (int k = NSTATE; k < 8; ++k) Zr[k * CPR + c] = 0.f;   // K padding
        float zn[NSTATE];
        #pragma unroll
        for (int i = 0; i < NSTATE; ++i) {
            float s = fr[(size_t)c * 8 + i];
            #pragma unroll
            for (int j = 0; j < NSTATE; ++j) s = fmaf(P[i * NSTATE + j], z[j], s);
            zn[i] = s;
        }
        #pragma unroll
        for (int i = 0; i < NSTATE; ++i) z[i] = zn[i];
    }
}

// ---------------- kernel 5: WMMA correction GEMM + final rescale ----------------
// dOut tile(16 time x 16 chunks) += H(16x5) * Zstart(5x16), then *= xmax.
__global__ void __launch_bounds__(256) k_phase3(float* __restrict__ out,
                                                const float* __restrict__ ws) {
    const int lane = threadIdx.x & 31;
    const int wave = threadIdx.x >> 5;
    const unsigned TPR = (CPR / 16) * (CHUNK_L / 16);       // 32 * 128 = 4096 tiles/row
    unsigned tile = blockIdx.x * 8u + (unsigned)wave;       // 65536 tiles total
    unsigned row  = tile / TPR;
    unsigned rem  = tile % TPR;
    unsigned cg   = rem / (CHUNK_L / 16);                   // chunk group 0..31
    unsigned tg   = rem % (CHUNK_L / 16);                   // time group 0..127
    int t0 = (int)tg * 16, c0 = (int)cg * 16;

    const int half = lane >> 4;       // 0: K={0,1}/{4,5}, 1: K={2,3}/{6,7}
    const int li   = lane & 15;
    const int kb   = half * 2;

    const float* H  = ws + OFF_H;
    const float* Zr = ws + OFF_Z + (size_t)row * 8 * CPR;
    float xmax = __uint_as_float(((const unsigned*)ws)[OFF_MAXBITS]);

    // A (16x4 f32): lane half selects K pair; rows = time within tile
    const float* hrow = H + (size_t)(t0 + li) * 8;
    v2f a0 = { hrow[kb],     hrow[kb + 1] };     // K 0..3
    v2f a1 = { hrow[kb + 4], hrow[kb + 5] };     // K 4..7 (cols 5..7 are zero pad)

    // B (4x16 f32): columns = chunk within group
    int c = c0 + li;
    v2f b0 = { Zr[(kb)     * CPR + c], Zr[(kb + 1) * CPR + c] };
    v2f b1 = { Zr[(kb + 4) * CPR + c], Zr[(kb + 5) * CPR + c] };

    v8f acc = {};
    acc = __builtin_amdgcn_wmma_f32_16x16x4_f32(false, a0, false, b0, (short)0, acc, false, false);
    acc = __builtin_amdgcn_wmma_f32_16x16x4_f32(false, a1, false, b1, (short)0, acc, false, false);

    // D layout: lane holds N=li, M = v + 8*half for VGPR v -> 8 consecutive times
    float* op = out + (size_t)row * T_LEN + (size_t)c * CHUNK_L + t0 + half * 8;
    #pragma unroll
    for (int v = 0; v < 8; ++v)
        op[v] = (op[v] + acc[v]) * xmax;        // y = xmax * (y_zerostate + correction)
}

// ---------------- launcher ----------------
extern "C" void kernel_launch(void* const* d_in, const int* in_sizes, int n_in,
                              void* d_out, int out_size, void* d_ws, size_t ws_size,
                              hipStream_t stream) {
    const float* x  = (const float*)d_in[0];
    const float* bc = (const float*)d_in[1];
    const float* ac = (const float*)d_in[2];
    float*    out = (float*)d_out;
    float*    ws  = (float*)d_ws;
    unsigned* wsu = (unsigned*)d_ws;
    const int n = BATCH * (int)T_LEN;

    k_init<<<1, 64, 0, stream>>>(wsu);
    k_absmax<<<2048, 256, 0, stream>>>(x, wsu, n);
    k_prep<<<1, 64, 0, stream>>>(bc, ac, ws);
    k_phase1<<<NCHUNK / 256, 256, 0, stream>>>(x, bc, ac, out, ws);
    k_phase2<<<1, 32, 0, stream>>>(ws);
    const unsigned tiles = BATCH * (CPR / 16) * (CHUNK_L / 16);  // 65536
    k_phase3<<<tiles / 8, 256, 0, stream>>>(out, ws);
}